// BIGCN_CCFD_84911503442514
// MI455X (gfx1250) — compile-verified
//
#include <hip/hip_runtime.h>
#include <math.h>

typedef float v2f __attribute__((ext_vector_type(2)));
typedef float v8f __attribute__((ext_vector_type(8)));

constexpr int BATCH = 64;
constexpr int NNODE = 512;
constexpr int NIN   = 768;
constexpr int ND4   = 256;
constexpr int BN    = BATCH * NNODE;   // 32768
constexpr int MT    = 64;              // rows per block (4 x 16 sub-tiles / wave)

// d_out flat layout (floats)
constexpr size_t OFF_PREDS = 0;
constexpr size_t OFF_PXC   = 128;
constexpr size_t OFF_PXS   = 256;
constexpr size_t OFF_XC    = 384;
constexpr size_t OFF_XS    = 384 + 16384;
constexpr size_t OFF_XREC  = 384 + 2 * 16384;             // 33152
constexpr size_t OFF_DGATE = OFF_XREC + (size_t)BN * NIN; // 25198976

__device__ __forceinline__ float lrelu(float z) { return z >= 0.0f ? z : 0.01f * z; }

// Shared inner product step: one B-operand pair feeds 4 WMMAs (4 row sub-tiles)
#define WMMA_CHUNK(SA, BX, BY)                                                  \
  {                                                                             \
    v2f bb; bb.x = (BX); bb.y = (BY);                                           \
    _Pragma("unroll")                                                           \
    for (int s = 0; s < 4; ++s) {                                               \
      v2f a;                                                                    \
      a.x = (SA)[(s * 16 + l16) * 32 + k4 + 2 * half];                          \
      a.y = (SA)[(s * 16 + l16) * 32 + k4 + 2 * half + 1];                      \
      acc[s] = __builtin_amdgcn_wmma_f32_16x16x4_f32(false, a, false, bb,       \
                                                     (short)0, acc[s], false,   \
                                                     false);                    \
    }                                                                           \
  }

// ---------------------------------------------------------------------------
// Kernel 1: h = LN(lrelu(x @ emb_W)); maskf = (sum(LN row) == 0)
// M=32768, K=768, Ncols=64. 64-row tile, 4 waves (16 cols each).
// ---------------------------------------------------------------------------
__global__ __launch_bounds__(128)
void emb_ln_kernel(const float* __restrict__ x, const float* __restrict__ W,
                   float* __restrict__ h, float* __restrict__ maskf)
{
  __shared__ __align__(16) float sA[MT * 32];
  __shared__ float sZ[MT][65];
  __shared__ float sM[MT], sR[MT];
  const int t = threadIdx.x;
  const int wave = t >> 5, lane = t & 31, half = lane >> 4, l16 = lane & 15;
  const int row0 = blockIdx.x * MT;
  const int colbase = wave * 16;
  v8f acc[4] = {};
  for (int kb = 0; kb < NIN; kb += 32) {
#pragma unroll
    for (int i = 0; i < 4; ++i) {
      int fi = t + 128 * i, r = fi >> 3, kk = (fi & 7) * 4;
      *(float4*)(sA + r * 32 + kk) =
          *(const float4*)(x + (size_t)(row0 + r) * NIN + kb + kk);
    }
    __syncthreads();
#pragma unroll
    for (int k4 = 0; k4 < 32; k4 += 4) {
      int kg = kb + k4 + 2 * half;
      WMMA_CHUNK(sA, W[(size_t)kg * 64 + colbase + l16],
                     W[(size_t)(kg + 1) * 64 + colbase + l16]);
    }
    __syncthreads();
  }
#pragma unroll
  for (int s = 0; s < 4; ++s)
#pragma unroll
    for (int v = 0; v < 8; ++v)
      sZ[s * 16 + v + 8 * half][colbase + l16] = lrelu(acc[s][v]);
  __syncthreads();
  if (t < MT) {
    float sum = 0.f;
    for (int c = 0; c < 64; ++c) sum += sZ[t][c];
    float m = sum * (1.0f / 64.0f);
    float vv = 0.f;
    for (int c = 0; c < 64; ++c) { float d = sZ[t][c] - m; vv += d * d; }
    sM[t] = m;
    sR[t] = rsqrtf(vv * (1.0f / 64.0f) + 1e-5f);
  }
  __syncthreads();
  {
    int rr = t >> 1, c0 = (t & 1) * 32;
    for (int i = 0; i < 32; ++i)
      h[(size_t)(row0 + rr) * 64 + c0 + i] = (sZ[rr][c0 + i] - sM[rr]) * sR[rr];
  }
  if (t < MT) {
    float sum = 0.f;
    for (int c = 0; c < 64; ++c) sum += (sZ[t][c] - sM[t]) * sR[t];
    maskf[row0 + t] = (sum == 0.0f) ? 1.0f : 0.0f;
  }
}

// ---------------------------------------------------------------------------
// Kernel 2: out[r,0:64] = Xin[r,:] @ W (K x 64)
//   mode bit0: synthesize Xin[r,k] = maskf[r]*xall[b,k]*npost[n,k]  (K=256)
//   mode bit1: out += (1-maskf[r]) * r1vec[b,:]
// ---------------------------------------------------------------------------
__global__ __launch_bounds__(128)
void gemm_rows_kernel(const float* __restrict__ Xin, const float* __restrict__ W,
                      int K, const float* __restrict__ maskf,
                      const float* __restrict__ xall, const float* __restrict__ npost,
                      const float* __restrict__ r1vec, float* __restrict__ out,
                      int mode)
{
  __shared__ __align__(16) float sA[MT * 32];
  const int t = threadIdx.x;
  const int wave = t >> 5, lane = t & 31, half = lane >> 4, l16 = lane & 15;
  const int row0 = blockIdx.x * MT;
  const int colbase = wave * 16;
  v8f acc[4] = {};
  for (int kb = 0; kb < K; kb += 32) {
#pragma unroll
    for (int i = 0; i < 4; ++i) {
      int fi = t + 128 * i, r = fi >> 3, kk = (fi & 7) * 4;
      int grow = row0 + r;
      if (mode & 1) {
        float mf = maskf[grow];
        int b = grow >> 9, n = grow & 511;
        for (int j = 0; j < 4; ++j)
          sA[r * 32 + kk + j] = mf * xall[b * ND4 + kb + kk + j] *
                                npost[(size_t)n * ND4 + kb + kk + j];
      } else {
        *(float4*)(sA + r * 32 + kk) =
            *(const float4*)(Xin + (size_t)grow * K + kb + kk);
      }
    }
    __syncthreads();
#pragma unroll
    for (int k4 = 0; k4 < 32; k4 += 4) {
      int kg = kb + k4 + 2 * half;
      WMMA_CHUNK(sA, W[(size_t)kg * 64 + colbase + l16],
                     W[(size_t)(kg + 1) * 64 + colbase + l16]);
    }
    __syncthreads();
  }
  const int cc = colbase + l16;
#pragma unroll
  for (int s = 0; s < 4; ++s)
#pragma unroll
    for (int v = 0; v < 8; ++v) {
      int r = row0 + s * 16 + v + 8 * half;
      float o = acc[s][v];
      if (mode & 2) o += (1.0f - maskf[r]) * r1vec[(r >> 9) * 64 + cc];
      out[(size_t)r * 64 + cc] = o;
    }
}

// ---------------------------------------------------------------------------
// Kernel 3: Z[b] = op(A[b]) @ Y[b] + bias;  epilogue lrelu (epi=0) or lrelu+LN
//   amode 0: A ; 1: A^T ; 2: sym = (A^T==1 ? 1 : A)
// ---------------------------------------------------------------------------
__global__ __launch_bounds__(128)
void ax_kernel(const float* __restrict__ A, const float* __restrict__ Y,
               const float* __restrict__ bias, float* __restrict__ out,
               int amode, int epi)
{
  __shared__ __align__(16) float sA[MT * 32];
  __shared__ float sZ[MT][65];
  __shared__ float sM[MT], sR[MT];
  const int t = threadIdx.x;
  const int wave = t >> 5, lane = t & 31, half = lane >> 4, l16 = lane & 15;
  const int row0 = blockIdx.x * MT;
  const int b = blockIdx.y;
  const float* Ab = A + (size_t)b * NNODE * NNODE;
  const float* Yb = Y + (size_t)b * NNODE * 64;
  const int colbase = wave * 16;
  v8f acc[4] = {};
  for (int kb = 0; kb < NNODE; kb += 32) {
#pragma unroll
    for (int i = 0; i < 4; ++i) {
      int fi = t + 128 * i, r = fi >> 3, kk = (fi & 7) * 4;
      int grow = row0 + r;
      if (amode == 0) {
        *(float4*)(sA + r * 32 + kk) =
            *(const float4*)(Ab + (size_t)grow * NNODE + kb + kk);
      } else if (amode == 1) {
        for (int j = 0; j < 4; ++j)
          sA[r * 32 + kk + j] = Ab[(size_t)(kb + kk + j) * NNODE + grow];
      } else {
        for (int j = 0; j < 4; ++j) {
          float av = Ab[(size_t)grow * NNODE + kb + kk + j];
          float at = Ab[(size_t)(kb + kk + j) * NNODE + grow];
          sA[r * 32 + kk + j] = (at == 1.0f) ? 1.0f : av;
        }
      }
    }
    __syncthreads();
#pragma unroll
    for (int k4 = 0; k4 < 32; k4 += 4) {
      int kg = kb + k4 + 2 * half;
      WMMA_CHUNK(sA, Yb[(size_t)kg * 64 + colbase + l16],
                     Yb[(size_t)(kg + 1) * 64 + colbase + l16]);
    }
    __syncthreads();
  }
  const int cc = colbase + l16;
  const float bv = bias[cc];
  if (epi == 0) {
#pragma unroll
    for (int s = 0; s < 4; ++s)
#pragma unroll
      for (int v = 0; v < 8; ++v) {
        int r = row0 + s * 16 + v + 8 * half;
        out[((size_t)b * NNODE + r) * 64 + cc] = lrelu(acc[s][v] + bv);
      }
  } else {
#pragma unroll
    for (int s = 0; s < 4; ++s)
#pragma unroll
      for (int v = 0; v < 8; ++v)
        sZ[s * 16 + v + 8 * half][cc] = lrelu(acc[s][v] + bv);
    __syncthreads();
    if (t < MT) {
      float sum = 0.f;
      for (int c = 0; c < 64; ++c) sum += sZ[t][c];
      float m = sum * (1.0f / 64.0f);
      float vv = 0.f;
      for (int c = 0; c < 64; ++c) { float d = sZ[t][c] - m; vv += d * d; }
      sM[t] = m;
      sR[t] = rsqrtf(vv * (1.0f / 64.0f) + 1e-5f);
    }
    __syncthreads();
    int rr = t >> 1, c0 = (t & 1) * 32;
    for (int i = 0; i < 32; ++i)
      out[((size_t)b * NNODE + row0 + rr) * 64 + c0 + i] =
          (sZ[rr][c0 + i] - sM[rr]) * sR[rr];
  }
}

// ---------------------------------------------------------------------------
// r2vec[b,j] = x1[b,0,:] @ W2[64: ,j]   (rank-1 root term of concat GEMM)
// ---------------------------------------------------------------------------
__global__ void r2vec_kernel(const float* __restrict__ Xin, const float* __restrict__ Wb,
                             int F, const float* __restrict__ maskf,
                             const float* __restrict__ xall, const float* __restrict__ npost,
                             float* __restrict__ r2, int rec)
{
  int b = blockIdx.x, j = threadIdx.x;  // 64 threads
  float s = 0.f;
  if (rec) {
    float mf = maskf[b * NNODE];
    for (int k = 0; k < F; ++k)
      s += mf * xall[b * ND4 + k] * npost[k] * Wb[k * 64 + j];
  } else {
    const float* x0 = Xin + (size_t)b * NNODE * 64;
    for (int k = 0; k < F; ++k) s += x0[k] * Wb[k * 64 + j];
  }
  r2[b * 64 + j] = s;
}

// ---------------------------------------------------------------------------
// Pooled branch output: [mean_n H2 , H1[b,0,:]*frac_valid]
// ---------------------------------------------------------------------------
__global__ __launch_bounds__(128)
void pool_kernel(const float* __restrict__ H2, const float* __restrict__ H1,
                 const float* __restrict__ maskf, float* __restrict__ out)
{
  __shared__ float red[128];
  int b = blockIdx.x, t = threadIdx.x;
  float s = 0.f;
  for (int n = t; n < NNODE; n += 128) s += maskf[b * NNODE + n];
  red[t] = s;
  __syncthreads();
  for (int o = 64; o > 0; o >>= 1) {
    if (t < o) red[t] += red[t + o];
    __syncthreads();
  }
  float frac = ((float)NNODE - red[0]) * (1.0f / (float)NNODE);
  if (t < 64) {
    float a = 0.f;
    for (int n = 0; n < NNODE; ++n) a += H2[((size_t)b * NNODE + n) * 64 + t];
    out[b * 128 + t] = a * (1.0f / (float)NNODE);
  } else {
    out[b * 128 + t] = H1[(size_t)b * NNODE * 64 + (t - 64)] * frac;
  }
}

// ---------------------------------------------------------------------------
// Head: xf=concat(BU,TD); dgate/xc/xs/preds; writes outputs + xall to ws
// ---------------------------------------------------------------------------
__global__ __launch_bounds__(256)
void head_kernel(const float* __restrict__ BU, const float* __restrict__ TD,
                 const float* __restrict__ dgW, const float* __restrict__ dgb,
                 const float* __restrict__ caW, const float* __restrict__ cab,
                 const float* __restrict__ ccW, const float* __restrict__ ccb,
                 float* __restrict__ dout, float* __restrict__ xall_ws)
{
  __shared__ float xf[256], xcS[256], xaS[256];
  int b = blockIdx.x, t = threadIdx.x;
  xf[t] = (t < 128) ? BU[b * 128 + t] : TD[b * 128 + t - 128];
  __syncthreads();
  float a = 0.f;
  for (int k = 0; k < 256; ++k) a += xf[k] * dgW[k * 256 + t];
  float g = 1.0f / (1.0f + expf(-(a + dgb[t])));
  float xv = xf[t];
  float xc = xv * g, xs = xv * (1.0f - g);
  float xa = xc + xs;
  dout[OFF_DGATE + (size_t)b * 256 + t] = g;
  dout[OFF_XC + (size_t)b * 256 + t] = xc;
  dout[OFF_XS + (size_t)b * 256 + t] = xs;
  xall_ws[b * 256 + t] = xa;
  xcS[t] = xc;
  xaS[t] = xa;
  __syncthreads();
  if (t < 2) {
    float p = 0.f, pc = 0.f;
    for (int k = 0; k < 256; ++k) {
      p += xaS[k] * caW[k * 2 + t];
      pc += xcS[k] * ccW[k * 2 + t];
    }
    dout[OFF_PREDS + b * 2 + t] = p + cab[t];
    dout[OFF_PXC + b * 2 + t] = pc + ccb[t];
    dout[OFF_PXS + b * 2 + t] = p + cab[t];  // stop_gradient is identity in fwd
  }
}

// ---------------------------------------------------------------------------
// rvec[b,j] = H1[b,0,:] @ fc_W[64: , j]   (rank-1 root2 term of final GEMM)
// ---------------------------------------------------------------------------
__global__ __launch_bounds__(128)
void rvec_kernel(const float* __restrict__ H1, const float* __restrict__ fcB,
                 float* __restrict__ rvec)
{
  int b = blockIdx.x, j = blockIdx.y * 128 + threadIdx.x;  // 768 cols
  const float* h0 = H1 + (size_t)b * NNODE * 64;
  float s = 0.f;
  for (int k = 0; k < 64; ++k) s += h0[k] * fcB[k * NIN + j];
  rvec[b * NIN + j] = s;
}

// ---------------------------------------------------------------------------
// x_rec = H2 @ fc_W[:64] + (1-mask)*rvec ;  M=32768, K=64, N=768
// ---------------------------------------------------------------------------
__global__ __launch_bounds__(128)
void xrec_kernel(const float* __restrict__ H2, const float* __restrict__ fcW,
                 const float* __restrict__ maskf, const float* __restrict__ rvec,
                 float* __restrict__ out)
{
  __shared__ __align__(16) float sA[MT * 32];
  const int t = threadIdx.x;
  const int wave = t >> 5, lane = t & 31, half = lane >> 4, l16 = lane & 15;
  const int row0 = blockIdx.x * MT;
  const int colg = blockIdx.y * 64;
  const int colbase = wave * 16;
  v8f acc[4] = {};
  for (int kb = 0; kb < 64; kb += 32) {
#pragma unroll
    for (int i = 0; i < 4; ++i) {
      int fi = t + 128 * i, r = fi >> 3, kk = (fi & 7) * 4;
      *(float4*)(sA + r * 32 + kk) =
          *(const float4*)(H2 + (size_t)(row0 + r) * 64 + kb + kk);
    }
    __syncthreads();
#pragma unroll
    for (int k4 = 0; k4 < 32; k4 += 4) {
      int kg = kb + k4 + 2 * half;
      WMMA_CHUNK(sA, fcW[(size_t)kg * NIN + colg + colbase + l16],
                     fcW[(size_t)(kg + 1) * NIN + colg + colbase + l16]);
    }
    __syncthreads();
  }
  const int cc = colg + colbase + l16;
#pragma unroll
  for (int s = 0; s < 4; ++s)
#pragma unroll
    for (int v = 0; v < 8; ++v) {
      int r = row0 + s * 16 + v + 8 * half;
      int b = r >> 9;
      out[(size_t)r * NIN + cc] = acc[s][v] + (1.0f - maskf[r]) * rvec[b * NIN + cc];
    }
}

// ---------------------------------------------------------------------------
extern "C" void kernel_launch(void* const* d_in, const int* in_sizes, int n_in,
                              void* d_out, int out_size, void* d_ws, size_t ws_size,
                              hipStream_t stream)
{
  const float* x     = (const float*)d_in[0];
  const float* A     = (const float*)d_in[1];
  const float* emb_W = (const float*)d_in[2];
  const float* td_W1 = (const float*)d_in[3];
  const float* td_b1 = (const float*)d_in[4];
  const float* td_W2 = (const float*)d_in[5];
  const float* td_b2 = (const float*)d_in[6];
  const float* bu_W1 = (const float*)d_in[7];
  const float* bu_b1 = (const float*)d_in[8];
  const float* bu_W2 = (const float*)d_in[9];
  const float* bu_b2 = (const float*)d_in[10];
  const float* dg_W  = (const float*)d_in[11];
  const float* dg_b  = (const float*)d_in[12];
  const float* ca_W  = (const float*)d_in[13];
  const float* ca_b  = (const float*)d_in[14];
  const float* cc_W  = (const float*)d_in[15];
  const float* cc_b  = (const float*)d_in[16];
  const float* npost = (const float*)d_in[17];
  const float* rc_W1 = (const float*)d_in[18];
  const float* rc_b1 = (const float*)d_in[19];
  const float* rc_W2 = (const float*)d_in[20];
  const float* rc_b2 = (const float*)d_in[21];
  const float* fc_W  = (const float*)d_in[22];

  float* ws = (float*)d_ws;
  float* h     = ws;                       // 32768*64
  float* maskf = h + (size_t)BN * 64;      // 32768
  float* Y     = maskf + BN;               // 32768*64
  float* H1    = Y + (size_t)BN * 64;      // 32768*64
  float* H2    = H1 + (size_t)BN * 64;     // 32768*64
  float* TDp   = H2 + (size_t)BN * 64;     // 64*128
  float* BUp   = TDp + BATCH * 128;        // 64*128
  float* xall  = BUp + BATCH * 128;        // 64*256
  float* r2    = xall + BATCH * ND4;       // 64*64
  float* rvec  = r2 + BATCH * 64;          // 64*768
  float* dout  = (float*)d_out;

  const dim3 blk128(128), blkAX(NNODE / MT, BATCH);

  // ---- embedding + LN + mask
  emb_ln_kernel<<<BN / MT, blk128, 0, stream>>>(x, emb_W, h, maskf);

  // ---- TD branch (A)
  gemm_rows_kernel<<<BN / MT, blk128, 0, stream>>>(h, td_W1, 64, maskf, nullptr, nullptr, nullptr, Y, 0);
  ax_kernel<<<blkAX, blk128, 0, stream>>>(A, Y, td_b1, H1, 0, 1);
  r2vec_kernel<<<BATCH, 64, 0, stream>>>(h, td_W2 + 64 * 64, 64, maskf, nullptr, nullptr, r2, 0);
  gemm_rows_kernel<<<BN / MT, blk128, 0, stream>>>(H1, td_W2, 64, maskf, nullptr, nullptr, r2, Y, 2);
  ax_kernel<<<blkAX, blk128, 0, stream>>>(A, Y, td_b2, H2, 0, 0);
  pool_kernel<<<BATCH, blk128, 0, stream>>>(H2, H1, maskf, TDp);

  // ---- BU branch (A^T)
  gemm_rows_kernel<<<BN / MT, blk128, 0, stream>>>(h, bu_W1, 64, maskf, nullptr, nullptr, nullptr, Y, 0);
  ax_kernel<<<blkAX, blk128, 0, stream>>>(A, Y, bu_b1, H1, 1, 1);
  r2vec_kernel<<<BATCH, 64, 0, stream>>>(h, bu_W2 + 64 * 64, 64, maskf, nullptr, nullptr, r2, 0);
  gemm_rows_kernel<<<BN / MT, blk128, 0, stream>>>(H1, bu_W2, 64, maskf, nullptr, nullptr, r2, Y, 2);
  ax_kernel<<<blkAX, blk128, 0, stream>>>(A, Y, bu_b2, H2, 1, 0);
  pool_kernel<<<BATCH, blk128, 0, stream>>>(H2, H1, maskf, BUp);

  // ---- gating head
  head_kernel<<<BATCH, 256, 0, stream>>>(BUp, TDp, dg_W, dg_b, ca_W, ca_b, cc_W, cc_b, dout, xall);

  // ---- reconstruction branch (A_sym), recx synthesized on the fly
  gemm_rows_kernel<<<BN / MT, blk128, 0, stream>>>(nullptr, rc_W1, 256, maskf, xall, npost, nullptr, Y, 1);
  ax_kernel<<<blkAX, blk128, 0, stream>>>(A, Y, rc_b1, H1, 2, 1);
  r2vec_kernel<<<BATCH, 64, 0, stream>>>(nullptr, rc_W2 + 64 * 64, 256, maskf, xall, npost, r2, 1);
  gemm_rows_kernel<<<BN / MT, blk128, 0, stream>>>(H1, rc_W2, 64, maskf, nullptr, nullptr, r2, Y, 2);
  ax_kernel<<<blkAX, blk128, 0, stream>>>(A, Y, rc_b2, H2, 2, 0);
  rvec_kernel<<<dim3(BATCH, NIN / 128), blk128, 0, stream>>>(H1, fc_W + 64 * NIN, rvec);
  xrec_kernel<<<dim3(BN / MT, NIN / 64), blk128, 0, stream>>>(H2, fc_W, maskf, rvec, dout + OFF_XREC);
}